// AttentionPool_28595892256780
// MI455X (gfx1250) — compile-verified
//
#include <hip/hip_runtime.h>
#include <hip/hip_bf16.h>
#include <cstdint>

typedef __attribute__((ext_vector_type(2))) float v2f;
typedef __attribute__((ext_vector_type(8))) float v8f;

#define WAVES_PER_BLOCK 8

// ---------------------------------------------------------------------------
// Kernel A: logits[n] = dot(x[n, 0:128], W) + b   via V_WMMA_F32_16X16X4_F32
//
// One 16-node tile per wave.
//   A (16x4 fp32, documented layout): lane l holds row M = l%16,
//     VGPR0 = x[row, kb + 2*(l/16)], VGPR1 = x[row, kb + 2*(l/16) + 1]
//   B (4x16 fp32): W chunk broadcast across all 16 columns -> every column of
//     the 16x16 D accumulator carries the same logit (robust to B's N-map).
// Full tiles take an unguarded path (no exec-mask fences -> loads pipeline);
// only the final partial tile (N%16 != 0) takes the guarded path.
// Two accumulators halve the WMMA dependency chain.
// ---------------------------------------------------------------------------
template <bool FULL>
__device__ __forceinline__ void gate_tile(
    const float* __restrict__ x, const float* __restrict__ ldsW,
    float* __restrict__ logits, float bv, long long n0, int lane, int N)
{
    const int m = lane & 15;   // row within tile
    const int h = lane >> 4;   // lane half selects K pair
    const long long row = n0 + m;
    const bool row_ok = FULL || (row < (long long)N);
    const float* xr = x + row * 128 + 2 * h;
    const float* wr = ldsW + 2 * h;

    v8f c0 = {};
    v8f c1 = {};
#pragma unroll
    for (int kb = 0; kb < 128; kb += 8) {
        v2f a0, a1, w0, w1;
        if (FULL) {
            a0.x = xr[kb];     a0.y = xr[kb + 1];
            a1.x = xr[kb + 4]; a1.y = xr[kb + 5];
        } else {
            if (row_ok) {
                a0.x = xr[kb];     a0.y = xr[kb + 1];
                a1.x = xr[kb + 4]; a1.y = xr[kb + 5];
            } else {
                a0.x = 0.0f; a0.y = 0.0f; a1.x = 0.0f; a1.y = 0.0f;
            }
        }
        w0.x = wr[kb];     w0.y = wr[kb + 1];
        w1.x = wr[kb + 4]; w1.y = wr[kb + 5];
        c0 = __builtin_amdgcn_wmma_f32_16x16x4_f32(false, a0, false, w0,
                                                   (short)0, c0, false, false);
        c1 = __builtin_amdgcn_wmma_f32_16x16x4_f32(false, a1, false, w1,
                                                   (short)0, c1, false, false);
    }

    // Extraction (documented C/D layout): lane with l%16==0 holds rows
    //   M = v + 8*(l/16) in accumulator element v.
    if (m == 0) {
        const long long nb = n0 + 8 * h;
#pragma unroll
        for (int v = 0; v < 8; ++v) {
            if (FULL || (nb + v < (long long)N))
                logits[nb + v] = c0[v] + c1[v] + bv;
        }
    }
}

__global__ __launch_bounds__(256) void gate_logits_wmma(
    const float* __restrict__ x, const float* __restrict__ W,
    const float* __restrict__ bias, float* __restrict__ logits, int N)
{
    __shared__ float ldsW[128];
    const int tid = threadIdx.x;
    if (tid < 128) ldsW[tid] = W[tid];
    __syncthreads();

    const int wave = tid >> 5;
    const int lane = tid & 31;
    const long long tile = (long long)blockIdx.x * WAVES_PER_BLOCK + wave;
    const long long n0 = tile * 16;
    if (n0 >= N) return;

    const float bv = bias[0];
    if (n0 + 16 <= (long long)N) {
        gate_tile<true >(x, ldsW, logits, bv, n0, lane, N);   // hot path
    } else {
        gate_tile<false>(x, ldsW, logits, bv, n0, lane, N);   // rare tail
    }
}

// ---------------------------------------------------------------------------
// Kernel B: one block (256 threads) per graph.
//   segment bounds via binary search on sorted batch[],
//   block-reduce max -> sum(exp) -> weighted accumulate with b128 loads
//   (each thread owns 4 columns; 8 rows/iter; unrolled x2 for 2 loads in
//   flight per wave), final 8-way combine through LDS.
// ---------------------------------------------------------------------------
__device__ __forceinline__ int lower_bound_i32(const int* __restrict__ a,
                                               int n, int key)
{
    int lo = 0, hi = n;
    while (lo < hi) {
        int mid = (lo + hi) >> 1;
        if (a[mid] < key) lo = mid + 1; else hi = mid;
    }
    return lo;
}

__global__ __launch_bounds__(256) void segment_pool(
    const float* __restrict__ x, const int* __restrict__ batch,
    const float* __restrict__ logits, float* __restrict__ out, int N)
{
    __shared__ float red[256];
    __shared__ float part[8][128];
    const int tid = threadIdx.x;
    const int g   = blockIdx.x;

    const int s = lower_bound_i32(batch, N, g);
    const int e = lower_bound_i32(batch, N, g + 1);

    // --- per-segment max ---
    float mx = -3.402823466e+38f;
    for (int i = s + tid; i < e; i += 256) mx = fmaxf(mx, logits[i]);
    red[tid] = mx;
    __syncthreads();
#pragma unroll
    for (int o = 128; o > 0; o >>= 1) {
        if (tid < o) red[tid] = fmaxf(red[tid], red[tid + o]);
        __syncthreads();
    }
    const float M = red[0];
    __syncthreads();

    // --- per-segment sum of exp (logits re-read is L2-hot) ---
    float ss = 0.0f;
    for (int i = s + tid; i < e; i += 256) ss += __expf(logits[i] - M);
    red[tid] = ss;
    __syncthreads();
#pragma unroll
    for (int o = 128; o > 0; o >>= 1) {
        if (tid < o) red[tid] += red[tid + o];
        __syncthreads();
    }
    const float invS = 1.0f / (red[0] + 1e-16f);
    __syncthreads();

    // --- weighted accumulate: b128 per thread, 8 rows / block / iter ---
    const int lane  = tid & 31;      // column group: cols [4*lane, 4*lane+4)
    const int rgrp  = tid >> 5;      // row subgroup 0..7
    const int col4  = lane * 4;

    float4 acc = make_float4(0.f, 0.f, 0.f, 0.f);
    int n = s + rgrp;
    for (; n + 8 < e; n += 16) {
        const float w0 = __expf(logits[n]     - M) * invS;
        const float w1 = __expf(logits[n + 8] - M) * invS;
        const float4 r0 = *(const float4*)(x + (long long)n       * 128 + col4);
        const float4 r1 = *(const float4*)(x + (long long)(n + 8) * 128 + col4);
        acc.x += w0 * r0.x + w1 * r1.x;
        acc.y += w0 * r0.y + w1 * r1.y;
        acc.z += w0 * r0.z + w1 * r1.z;
        acc.w += w0 * r0.w + w1 * r1.w;
    }
    if (n < e) {
        const float w0 = __expf(logits[n] - M) * invS;
        const float4 r0 = *(const float4*)(x + (long long)n * 128 + col4);
        acc.x += w0 * r0.x;
        acc.y += w0 * r0.y;
        acc.z += w0 * r0.z;
        acc.w += w0 * r0.w;
    }

    *(float4*)(&part[rgrp][col4]) = acc;
    __syncthreads();

    if (tid < 128) {
        float v = part[0][tid];
#pragma unroll
        for (int r = 1; r < 8; ++r) v += part[r][tid];
        out[(long long)g * 128 + tid] = v;
    }
}

// ---------------------------------------------------------------------------
// Host launch
// inputs: [0]=x (N*128 f32), [1]=batch (N int), [2]=W (128 f32),
//         [3]=b (1 f32), [4]=num_graphs (1 int)
// d_ws: N floats of logits scratch (4 MB).
// ---------------------------------------------------------------------------
extern "C" void kernel_launch(void* const* d_in, const int* in_sizes, int n_in,
                              void* d_out, int out_size, void* d_ws, size_t ws_size,
                              hipStream_t stream)
{
    const float* x     = (const float*)d_in[0];
    const int*   batch = (const int*)d_in[1];
    const float* W     = (const float*)d_in[2];
    const float* bias  = (const float*)d_in[3];

    const int N = in_sizes[1];           // number of nodes
    const int B = out_size / 128;        // number of graphs (D = 128)

    float* logits = (float*)d_ws;        // N floats

    const int tiles   = (N + 15) / 16;
    const int blocksA = (tiles + WAVES_PER_BLOCK - 1) / WAVES_PER_BLOCK;

    gate_logits_wmma<<<blocksA, 256, 0, stream>>>(x, W, bias, logits, N);
    segment_pool<<<B, 256, 0, stream>>>(x, batch, logits, (float*)d_out, N);
}